// MASK_MSE_LOSS_25658134626814
// MI455X (gfx1250) — compile-verified
//
#include <hip/hip_runtime.h>
#include <stdint.h>

#define TPB    256
#define CHUNK  4096
#define VEC    4
#define VITERS (CHUNK / (TPB * VEC))   // 4
#define NB0    2048    // level-0/1 bins (11 bits)
#define NB2    1024    // level-2 bins (10 bits)
#define NSAMP  16

typedef __attribute__((ext_vector_type(2))) float v2f;
typedef __attribute__((ext_vector_type(8))) float v8f;

// Wave32 sum reduction via V_WMMA_F32_16X16X4_F32 (fp32-exact, result broadcast).
__device__ __forceinline__ float waveReduceSumWMMA(float v) {
    v2f a;   a[0] = v;   a[1] = 0.0f;
    v2f one; one[0] = 1.0f; one[1] = 1.0f;
    v8f z = {};
    v8f d = __builtin_amdgcn_wmma_f32_16x16x4_f32(false, a, false, one, (short)0, z, false, false);
    v8f acc = {};
#pragma unroll
    for (int i = 0; i < 4; ++i) {
        v2f b; b[0] = d[2 * i]; b[1] = d[2 * i + 1];
        acc = __builtin_amdgcn_wmma_f32_16x16x4_f32(false, one, false, b, (short)0, acc, false, false);
    }
    return acc[0];
}

__global__ void k_zero(uint32_t* p, size_t n) {
    size_t i = (size_t)blockIdx.x * blockDim.x + threadIdx.x;
    if (i < n) p[i] = 0u;
}

// Streaming pass: b128 loads, scalars, level-0 histogram, wave-aggregated compaction.
__global__ void k_pass1(const float* __restrict__ x, const float* __restrict__ y,
                        const int* __restrict__ msk, int N, int bps,
                        float* __restrict__ negBuf, uint32_t* negCnt, uint32_t* numPos,
                        float* sumValid, float* sumPos,
                        uint32_t* cnt0, float* sum0, int useCompact) {
    __shared__ uint32_t hc[NB0];
    __shared__ float    hs[NB0];
    __shared__ float    stage[CHUNK];
    __shared__ uint32_t sNeg, sPosCnt, sBase;
    __shared__ float    sValid, sPos;

    int b = blockIdx.x / bps;
    int chunk = blockIdx.x % bps;
    int tid = threadIdx.x;
    int lane = tid & 31;

    for (int i = tid; i < NB0; i += TPB) { hc[i] = 0u; hs[i] = 0.0f; }
    if (tid == 0) { sNeg = 0u; sPosCnt = 0u; sValid = 0.0f; sPos = 0.0f; }
    __syncthreads();

    size_t sbase = (size_t)b * (size_t)N;
    float lv = 0.0f, lp = 0.0f;
    uint32_t lpc = 0u;

    for (int it = 0; it < VITERS; ++it) {
        int i0 = chunk * CHUNK + (it * TPB + tid) * VEC;
        float l[VEC]; int mm[VEC];
        if (i0 + VEC <= N) {
            float4 xv = *(const float4*)(x + sbase + i0);
            float4 yv = *(const float4*)(y + sbase + i0);
            int4   mv = *(const int4*)(msk + sbase + i0);
            float d0 = xv.x - yv.x, d1 = xv.y - yv.y, d2 = xv.z - yv.z, d3 = xv.w - yv.w;
            l[0] = d0 * d0; l[1] = d1 * d1; l[2] = d2 * d2; l[3] = d3 * d3;
            mm[0] = mv.x; mm[1] = mv.y; mm[2] = mv.z; mm[3] = mv.w;
        } else {
#pragma unroll
            for (int e = 0; e < VEC; ++e) {
                int i = i0 + e;
                if (i < N) {
                    float dv = x[sbase + i] - y[sbase + i];
                    l[e] = dv * dv; mm[e] = msk[sbase + i];
                } else { l[e] = 0.0f; mm[e] = 0; }
            }
        }
#pragma unroll
        for (int e = 0; e < VEC; ++e) {
            int m = mm[e]; float le = l[e];
            if (m != 0) lv += le;
            if (m == 2) { lp += le; lpc++; }
            bool neg = (m == 1);
            unsigned msk32 = (unsigned)__ballot(neg);     // convergent
            if (msk32) {
                int leader = __ffs(msk32) - 1;
                unsigned base = 0u;
                if (lane == leader) base = atomicAdd(&sNeg, (unsigned)__popc(msk32));
                base = __shfl(base, leader);
                if (neg) {
                    unsigned off = (unsigned)__popc(msk32 & ((1u << lane) - 1u));
                    if (useCompact) stage[base + off] = le;
                    uint32_t bits = __float_as_uint(le);  // le >= 0 -> monotone bits
                    uint32_t bin = bits >> 21;
                    atomicAdd(&hc[bin], 1u);
                    atomicAdd(&hs[bin], le);
                }
            }
        }
    }

    // Convergent point: EXEC all-1s -> legal WMMA.
    float wv = waveReduceSumWMMA(lv);
    float wp = waveReduceSumWMMA(lp);
    if (lane == 0) { atomicAdd(&sValid, wv); atomicAdd(&sPos, wp); }
    if (lpc) atomicAdd(&sPosCnt, lpc);
    __syncthreads();

    if (tid == 0) {
        atomicAdd(&sumValid[b], sValid);
        atomicAdd(&sumPos[b], sPos);
        if (sPosCnt) atomicAdd(&numPos[b], sPosCnt);
        sBase = atomicAdd(&negCnt[b], sNeg);
    }
    __syncthreads();

    if (useCompact) {
        for (uint32_t p = tid; p < sNeg; p += TPB)
            negBuf[sbase + (size_t)(sBase + p)] = stage[p];
    }
    for (int bin = tid; bin < NB0; bin += TPB) {
        if (hc[bin]) {
            atomicAdd(&cnt0[b * NB0 + bin], hc[bin]);
            atomicAdd(&sum0[b * NB0 + bin], hs[bin]);
        }
    }
}

// level==1: match bits>>21 == pref, bin = (bits>>10)&2047 (2048 bins)
// level==2: match bits>>10 == pref, bin = bits&1023       (1024 bins)
__global__ void k_refine(const float* __restrict__ x, const float* __restrict__ y,
                         const int* __restrict__ msk, int N, int bps,
                         const float* __restrict__ negBuf, const uint32_t* negCnt,
                         const uint32_t* done, const uint32_t* prefA,
                         uint32_t* cntH, float* sumH, int level, int useCompact) {
    __shared__ uint32_t hc[NB0];
    __shared__ float    hs[NB0];
    int b = blockIdx.x / bps;
    if (done[b]) return;                              // uniform per block
    int chunk = blockIdx.x % bps;
    int tid = threadIdx.x;
    int nb = (level == 1) ? NB0 : NB2;

    for (int i = tid; i < nb; i += TPB) { hc[i] = 0u; hs[i] = 0.0f; }
    __syncthreads();

    uint32_t pref = prefA[b];
    size_t sbase = (size_t)b * (size_t)N;
    int limit = useCompact ? (int)negCnt[b] : N;

    auto process = [&](float l) {
        uint32_t bits = __float_as_uint(l);
        if (level == 1) {
            if ((bits >> 21) == pref) {
                uint32_t bin = (bits >> 10) & 2047u;
                atomicAdd(&hc[bin], 1u); atomicAdd(&hs[bin], l);
            }
        } else {
            if ((bits >> 10) == pref) {
                uint32_t bin = bits & 1023u;
                atomicAdd(&hc[bin], 1u); atomicAdd(&hs[bin], l);
            }
        }
    };

    for (int it = 0; it < VITERS; ++it) {
        int i0 = chunk * CHUNK + (it * TPB + tid) * VEC;
        if (useCompact) {
            if (i0 + VEC <= limit) {
                float4 v = *(const float4*)(negBuf + sbase + i0);
                process(v.x); process(v.y); process(v.z); process(v.w);
            } else {
                for (int e = 0; e < VEC; ++e) {
                    int i = i0 + e;
                    if (i < limit) process(negBuf[sbase + i]);
                }
            }
        } else {
            if (i0 + VEC <= limit) {
                float4 xv = *(const float4*)(x + sbase + i0);
                float4 yv = *(const float4*)(y + sbase + i0);
                int4   mv = *(const int4*)(msk + sbase + i0);
                float d;
                d = xv.x - yv.x; if (mv.x == 1) process(d * d);
                d = xv.y - yv.y; if (mv.y == 1) process(d * d);
                d = xv.z - yv.z; if (mv.z == 1) process(d * d);
                d = xv.w - yv.w; if (mv.w == 1) process(d * d);
            } else {
                for (int e = 0; e < VEC; ++e) {
                    int i = i0 + e;
                    if (i < limit && msk[sbase + i] == 1) {
                        float dv = x[sbase + i] - y[sbase + i];
                        process(dv * dv);
                    }
                }
            }
        }
    }
    __syncthreads();
    for (int bin = tid; bin < nb; bin += TPB)
        if (hc[bin]) { atomicAdd(&cntH[b * nb + bin], hc[bin]); atomicAdd(&sumH[b * nb + bin], hs[bin]); }
}

// One block per sample; parallel suffix-scan select over nb bins.
__global__ void k_select(int level, int nb,
                         const uint32_t* __restrict__ cntH, const float* __restrict__ sumH,
                         const uint32_t* negCnt, const uint32_t* numPos,
                         const float* sumValid, const float* sumPos,
                         uint32_t* done, float* perSample,
                         uint32_t* prefix0, uint32_t* prefix01,
                         uint32_t* kRem, uint32_t* kArr, float* sumAbove) {
    __shared__ uint32_t pC[TPB];
    __shared__ float    pS[TPB];
    __shared__ uint32_t sK, sSimple;

    int s = blockIdx.x;
    int tid = threadIdx.x;

    if (tid == 0) {
        if (level == 0) {
            uint32_t np = numPos[s], nn = negCnt[s];
            int k = 3 * (int)np;                      // floor(np*3.0) exact
            if (k >= (int)nn || k <= 10) {            // use_simple branch
                done[s] = 1u; sSimple = 1u;
                uint32_t nall = np + nn;
                perSample[s] = sumValid[s] / (float)(nall ? nall : 1u);
            } else {
                done[s] = 0u; sSimple = 0u;
                kArr[s] = (uint32_t)k; sK = (uint32_t)k;
            }
        } else {
            sSimple = done[s];
            sK = kRem[s];
        }
    }
    __syncthreads();
    if (sSimple) return;                              // uniform

    const int SEG = nb / TPB;                         // 8 or 4
    uint32_t lc = 0u; float ls = 0.0f;
    for (int e = 0; e < SEG; ++e) {
        int bin = nb - 1 - (tid * SEG + e);           // reversed (descending value)
        lc += cntH[s * nb + bin];
        ls += sumH[s * nb + bin];
    }
    pC[tid] = lc; pS[tid] = ls;
    __syncthreads();
    for (int off = 1; off < TPB; off <<= 1) {         // Hillis-Steele inclusive scan
        uint32_t c = (tid >= off) ? pC[tid - off] : 0u;
        float    f = (tid >= off) ? pS[tid - off] : 0.0f;
        __syncthreads();
        pC[tid] += c; pS[tid] += f;
        __syncthreads();
    }
    uint32_t cum = (tid > 0) ? pC[tid - 1] : 0u;      // exclusive suffix count above my segment
    float    sAb = (tid > 0) ? pS[tid - 1] : 0.0f;
    uint32_t k = sK;

    if (cum < k && cum + lc >= k) {                   // unique crossing thread
        for (int e = 0; e < SEG; ++e) {
            int bin = nb - 1 - (tid * SEG + e);
            uint32_t c = cntH[s * nb + bin];
            if (cum + c >= k) {
                if (level == 0) {
                    prefix0[s] = (uint32_t)bin;
                    kRem[s] = k - cum;
                    sumAbove[s] = sAb;
                } else if (level == 1) {
                    prefix01[s] = (prefix0[s] << 11) | (uint32_t)bin;
                    kRem[s] = k - cum;
                    sumAbove[s] += sAb;
                } else {
                    // final level: bin holds the exact 32-bit pattern
                    uint32_t bits = (prefix01[s] << 10) | (uint32_t)bin;
                    float v = __uint_as_float(bits);
                    float top = sumAbove[s] + sAb + (float)(k - cum) * v;
                    uint32_t np = numPos[s];
                    float posMean = sumPos[s] / (float)(np ? np : 1u);
                    perSample[s] = posMean + top / (float)kArr[s];
                }
                break;
            }
            cum += c; sAb += sumH[s * nb + bin];
        }
    }
}

__global__ void k_final_mean(const float* __restrict__ perSample, float* __restrict__ out) {
    int tid = threadIdx.x;                            // 32 threads, one wave
    float v = (tid < NSAMP) ? perSample[tid] : 0.0f;
    float t = waveReduceSumWMMA(v);                   // EXEC all-1s
    if (tid == 0) out[0] = t / (float)NSAMP;
}

extern "C" void kernel_launch(void* const* d_in, const int* in_sizes, int n_in,
                              void* d_out, int out_size, void* d_ws, size_t ws_size,
                              hipStream_t stream) {
    const float* x = (const float*)d_in[0];
    const float* y = (const float*)d_in[1];
    const int*   m = (const int*)d_in[2];
    int total = in_sizes[0];
    const int B = NSAMP;
    int N = total / B;

    size_t metaWords = (size_t)B * (4 * NB0 + 2 * NB2 + 11);
    size_t metaBytes = metaWords * sizeof(uint32_t);
    size_t negBytes  = (size_t)total * sizeof(float);
    int useCompact = (ws_size >= negBytes + metaBytes) ? 1 : 0;

    char* ws = (char*)d_ws;
    float* negBuf = (float*)ws;
    uint32_t* meta = (uint32_t*)(ws + (useCompact ? negBytes : 0));
    uint32_t* metaBase = meta;

    uint32_t* cnt0 = meta;         meta += (size_t)B * NB0;
    float*    sum0 = (float*)meta; meta += (size_t)B * NB0;
    uint32_t* cnt1 = meta;         meta += (size_t)B * NB0;
    float*    sum1 = (float*)meta; meta += (size_t)B * NB0;
    uint32_t* cnt2 = meta;         meta += (size_t)B * NB2;
    float*    sum2 = (float*)meta; meta += (size_t)B * NB2;
    uint32_t* negCnt   = meta;  meta += B;
    uint32_t* numPos   = meta;  meta += B;
    float*    sumValid = (float*)meta; meta += B;
    float*    sumPos   = (float*)meta; meta += B;
    uint32_t* prefix0  = meta;  meta += B;
    uint32_t* prefix01 = meta;  meta += B;
    uint32_t* kRem     = meta;  meta += B;
    uint32_t* kArr     = meta;  meta += B;
    float*    sumAbove = (float*)meta; meta += B;
    uint32_t* done     = meta;  meta += B;
    float*    perSample = (float*)meta;

    int bps = (N + CHUNK - 1) / CHUNK;
    dim3 gz((unsigned)((metaWords + TPB - 1) / TPB));
    dim3 g1((unsigned)(B * bps));

    k_zero<<<gz, TPB, 0, stream>>>(metaBase, metaWords);
    k_pass1<<<g1, TPB, 0, stream>>>(x, y, m, N, bps, negBuf, negCnt, numPos,
                                    sumValid, sumPos, cnt0, sum0, useCompact);
    k_select<<<B, TPB, 0, stream>>>(0, NB0, cnt0, sum0, negCnt, numPos, sumValid, sumPos,
                                    done, perSample, prefix0, prefix01, kRem, kArr, sumAbove);
    k_refine<<<g1, TPB, 0, stream>>>(x, y, m, N, bps, negBuf, negCnt, done,
                                     prefix0, cnt1, sum1, 1, useCompact);
    k_select<<<B, TPB, 0, stream>>>(1, NB0, cnt1, sum1, negCnt, numPos, sumValid, sumPos,
                                    done, perSample, prefix0, prefix01, kRem, kArr, sumAbove);
    k_refine<<<g1, TPB, 0, stream>>>(x, y, m, N, bps, negBuf, negCnt, done,
                                     prefix01, cnt2, sum2, 2, useCompact);
    k_select<<<B, TPB, 0, stream>>>(2, NB2, cnt2, sum2, negCnt, numPos, sumValid, sumPos,
                                    done, perSample, prefix0, prefix01, kRem, kArr, sumAbove);
    k_final_mean<<<1, 32, 0, stream>>>(perSample, (float*)d_out);
}